// ShortConvolution_5016521802512
// MI455X (gfx1250) — compile-verified
//
#include <hip/hip_runtime.h>

// ---------------- configuration ----------------
#define BT    28                 // tokens per block tile
#define ROWS  (BT + 3)           // staged rows (window needs 3 history rows)
#define BDV   128                // float4 channel slices per block (512 floats)
#define BD    (BDV * 4)          // floats of channel per block
#define NTH   128                // threads per block (== BDV), 4 waves (wave32)
#define MAXCU 65                 // cu_seqlens entries cached in LDS

typedef float vf4 __attribute__((ext_vector_type(4)));  // native clang vector (for nontemporal b128 store)

__device__ __forceinline__ float silu_f(float a) {
    return a / (1.0f + __expf(-a));
}

// Async global -> LDS 16B copy (CDNA5, ASYNCcnt-tracked). laddr is the LDS
// byte offset (low 32 bits of a generic pointer to __shared__), boff the
// per-lane byte offset added to the 64-bit SGPR base (GVS mode).
__device__ __forceinline__ void async_ld_b128(unsigned laddr, unsigned boff,
                                              unsigned long long base) {
    asm volatile("global_load_async_to_lds_b128 %0, %1, %2"
                 :: "v"(laddr), "v"(boff), "s"(base) : "memory");
}

__device__ __forceinline__ void wait_async0() {
    asm volatile("s_wait_asynccnt 0x0" ::: "memory");
}

extern "C" __global__ __launch_bounds__(NTH)
void varlen_conv_silu(const float* __restrict__ x,
                      const float* __restrict__ wk,
                      const float* __restrict__ cache,
                      const int*   __restrict__ cu,
                      float*       __restrict__ y,
                      int T, int D, int nseq)
{
    __shared__ float tile[ROWS * BD];   // 31 * 2KB = 62KB
    __shared__ int   cuS[MAXCU];

    const int tid = (int)threadIdx.x;
    const int t0  = (int)blockIdx.x * BT;
    const int c0  = (int)blockIdx.y * BD;

    // cache cu_seqlens in LDS (small)
    const int ncu = nseq + 1;
    for (int i = tid; i < ncu && i < MAXCU; i += NTH) cuS[i] = cu[i];

    // ---- async stage: rows [t0-3, t0+BT) x channels [c0, c0+BD) into LDS ----
    const unsigned long long xbase = (unsigned long long)(const void*)x;
    for (int i = tid; i < ROWS * BDV; i += NTH) {   // 31 full-EXEC iterations
        const int r  = i >> 7;                      // / BDV
        const int cv = i & (BDV - 1);
        int g = t0 - 3 + r;
        g = g < 0 ? 0 : (g > T - 1 ? T - 1 : g);    // clamped rows are never consumed
        const unsigned boff  = ((unsigned)g * (unsigned)D + (unsigned)(c0 + cv * 4)) * 4u;
        const unsigned laddr = (unsigned)(unsigned long long)(const void*)&tile[r * BD + cv * 4];
        async_ld_b128(laddr, boff, xbase);
    }
    wait_async0();
    __syncthreads();

    const int* cp = (ncu <= MAXCU) ? (const int*)cuS : cu;

    // ---- compute: one float4 channel slice per thread, BT tokens ----
    const int cv = tid;                              // 0..BDV-1
    const float* wp = wk + (size_t)c0 + cv * 4;
    float4 wv[4];
    wv[0] = *(const float4*)(wp);
    wv[1] = *(const float4*)(wp + (size_t)D);
    wv[2] = *(const float4*)(wp + 2 * (size_t)D);
    wv[3] = *(const float4*)(wp + 3 * (size_t)D);

    const float4* t4 = (const float4*)tile;

    int s = 0;
    while (s + 1 < nseq && cp[s + 1] <= t0) ++s;
    int ss = cp[s];
    int se = cp[s + 1];

    float4 xm3 = t4[0 * BDV + cv];
    float4 xm2 = t4[1 * BDV + cv];
    float4 xm1 = t4[2 * BDV + cv];

    #pragma unroll 4
    for (int j = 0; j < BT; ++j) {
        const int t = t0 + j;
        if (t >= T) break;
        if (t >= se) { ++s; ss = cp[s]; se = cp[s + 1]; }
        const float4 xc = t4[(j + 3) * BDV + cv];
        float4 acc;
        if (t - ss >= 3) {
            // fast path: window entirely inside this sequence
            acc.x = wv[0].x * xm3.x + wv[1].x * xm2.x + wv[2].x * xm1.x + wv[3].x * xc.x;
            acc.y = wv[0].y * xm3.y + wv[1].y * xm2.y + wv[2].y * xm1.y + wv[3].y * xc.y;
            acc.z = wv[0].z * xm3.z + wv[1].z * xm2.z + wv[2].z * xm1.z + wv[3].z * xc.z;
            acc.w = wv[0].w * xm3.w + wv[1].w * xm2.w + wv[2].w * xm1.w + wv[3].w * xc.w;
        } else {
            // slow path: first 3 tokens of a sequence pull from cache
            acc.x = wv[3].x * xc.x;
            acc.y = wv[3].y * xc.y;
            acc.z = wv[3].z * xc.z;
            acc.w = wv[3].w * xc.w;
            #pragma unroll
            for (int k = 0; k < 3; ++k) {
                const int pos = t - 3 + k;
                float4 v;
                if (pos >= ss) {
                    v = t4[(pos - t0 + 3) * BDV + cv];
                } else {
                    const int cz = pos - ss + 3;        // cache_pos
                    if (cz >= 0) {
                        v = *(const float4*)(cache + ((size_t)s * 3 + cz) * D + c0 + cv * 4);
                    } else {
                        v = make_float4(0.f, 0.f, 0.f, 0.f);
                    }
                }
                acc.x += wv[k].x * v.x;
                acc.y += wv[k].y * v.y;
                acc.z += wv[k].z * v.z;
                acc.w += wv[k].w * v.w;
            }
        }
        vf4 o;
        o.x = silu_f(acc.x);
        o.y = silu_f(acc.y);
        o.z = silu_f(acc.z);
        o.w = silu_f(acc.w);
        // streaming store: y is write-once, keep it out of L2 so the x halo stays resident
        __builtin_nontemporal_store(o, (vf4*)(y + (size_t)t * D + c0 + cv * 4));
        xm3 = xm2; xm2 = xm1; xm1 = xc;
    }
}

// new_cache[n][j] = (idx = eos-3+j) >= bos ? x[idx] : cache[n][clip(j-(len-3),0,2)]
extern "C" __global__ void update_cache(const float* __restrict__ x,
                                        const float* __restrict__ cache,
                                        const int*   __restrict__ cu,
                                        float*       __restrict__ outc,
                                        int T, int D, int nseq)
{
    const int DV = D >> 2;
    const int total = nseq * 3 * DV;
    const int i = (int)(blockIdx.x * blockDim.x + threadIdx.x);
    if (i >= total) return;
    const int cv = i % DV;
    const int rj = i / DV;
    const int j  = rj % 3;
    const int n  = rj / 3;
    const int bos = cu[n];
    const int eos = cu[n + 1];
    const int len = eos - bos;
    const int idx = eos - 3 + j;
    float4 v;
    if (idx >= bos) {
        int ci = idx < 0 ? 0 : (idx > T - 1 ? T - 1 : idx);
        v = *(const float4*)(x + (size_t)ci * D + cv * 4);
    } else {
        int cidx = j - (len - 3);
        cidx = cidx < 0 ? 0 : (cidx > 2 ? 2 : cidx);
        v = *(const float4*)(cache + ((size_t)n * 3 + cidx) * D + cv * 4);
    }
    *(float4*)(outc + ((size_t)n * 3 + j) * D + cv * 4) = v;
}

extern "C" void kernel_launch(void* const* d_in, const int* in_sizes, int n_in,
                              void* d_out, int out_size, void* d_ws, size_t ws_size,
                              hipStream_t stream) {
    (void)n_in; (void)out_size; (void)d_ws; (void)ws_size;
    const float* x     = (const float*)d_in[0];
    const float* wk    = (const float*)d_in[1];
    const float* cache = (const float*)d_in[2];
    const int*   cu    = (const int*)d_in[3];

    const int nseq = in_sizes[3] - 1;          // N
    const int WD   = in_sizes[1];              // W*D
    const int WM1D = in_sizes[2] / nseq;       // (W-1)*D
    const int D    = WD - WM1D;                // 2048
    const int T    = in_sizes[0] / D;          // 16384

    float* y    = (float*)d_out;
    float* outc = y + (size_t)T * D;

    dim3 grid((T + BT - 1) / BT, D / BD);      // (586, 4)
    varlen_conv_silu<<<grid, NTH, 0, stream>>>(x, wk, cache, cu, y, T, D, nseq);

    const int total = nseq * 3 * (D / 4);
    update_cache<<<(total + 255) / 256, 256, 0, stream>>>(x, cache, cu, outc, T, D, nseq);
}